// GRU_GCN_50328426774822
// MI455X (gfx1250) — compile-verified
//
#include <hip/hip_runtime.h>
#include <math.h>

typedef __attribute__((ext_vector_type(2))) float v2f;
typedef __attribute__((ext_vector_type(8))) float v8f;

#define N_NODES 100000
#define N_EDGES 1000000
#define DIM 64
#define ND (N_NODES * DIM)
#define MTILES (N_NODES / 16)   // 6250 exactly

// ---------------------------------------------------------------------------
// Fused GEMM: out = inp @ W + b for one of {q,k,v,skip} selected by blockIdx.y
// One wave computes one 16-row M-tile of one 64-col output matrix using
// V_WMMA_F32_16X16X4_F32 (exact fp32, matches reference precision).
// A-frag layout (ISA 7.12.2, 32-bit A 16x4): lanes 0-15 -> K={k0,k0+1},
// lanes 16-31 -> K={k0+2,k0+3}; VGPR j holds K=k0+2*half+j.
// ---------------------------------------------------------------------------
__global__ void gemm_qkvs_kernel(const float* __restrict__ inp,
                                 const float* __restrict__ Wq, const float* __restrict__ bq,
                                 const float* __restrict__ Wk, const float* __restrict__ bk,
                                 const float* __restrict__ Wv, const float* __restrict__ bv,
                                 const float* __restrict__ Ws, const float* __restrict__ bs,
                                 float* __restrict__ Q, float* __restrict__ Kb,
                                 float* __restrict__ Vb, float* __restrict__ S) {
  int wave = threadIdx.x >> 5;
  int mtile = blockIdx.x * 8 + wave;
  if (mtile >= MTILES) return;   // whole-wave uniform exit: EXEC all-1 inside

  const float* W; const float* bias; float* out;
  switch (blockIdx.y) {
    case 0:  W = Wq; bias = bq; out = Q;  break;
    case 1:  W = Wk; bias = bk; out = Kb; break;
    case 2:  W = Wv; bias = bv; out = Vb; break;
    default: W = Ws; bias = bs; out = S;  break;
  }

  int lane = threadIdx.x & 31;
  int half = lane >> 4;   // 0: K pair {0,1}, 1: K pair {2,3}
  int l    = lane & 15;
  int row  = mtile * 16 + l;

  // Load all 16 A fragments (K = 0..63 in steps of 4) once; reuse for 4 N-tiles.
  const float* arow = inp + (size_t)row * DIM + 2 * half;
  v2f afrag[16];
#pragma unroll
  for (int ks = 0; ks < 16; ++ks) afrag[ks] = *(const v2f*)(arow + ks * 4);

#pragma unroll
  for (int nt = 0; nt < 4; ++nt) {
    int c0 = nt * 16;
    v8f acc = {0.f, 0.f, 0.f, 0.f, 0.f, 0.f, 0.f, 0.f};
#pragma unroll
    for (int ks = 0; ks < 16; ++ks) {
      int k0 = ks * 4 + 2 * half;
      v2f bfrag;
      bfrag.x = W[(k0 + 0) * DIM + c0 + l];
      bfrag.y = W[(k0 + 1) * DIM + c0 + l];
      acc = __builtin_amdgcn_wmma_f32_16x16x4_f32(false, afrag[ks], false, bfrag,
                                                  (short)0, acc, false, false);
    }
    float bb = bias[c0 + l];
#pragma unroll
    for (int v = 0; v < 8; ++v) {
      int r = mtile * 16 + v + 8 * half;   // C/D layout: VGPR v -> row v (+8 for hi lanes)
      out[(size_t)r * DIM + c0 + l] = acc[v] + bb;
    }
  }
}

// ---------------------------------------------------------------------------
// Attention edge phase
// ---------------------------------------------------------------------------
__global__ void init_seg_kernel(unsigned* __restrict__ mEnc, float* __restrict__ denom, int n) {
  int i = blockIdx.x * blockDim.x + threadIdx.x;
  if (i < n) { mEnc[i] = 0u; denom[i] = 0.f; }
}

__device__ __forceinline__ unsigned enc_f32(float f) {
  unsigned b = __float_as_uint(f);
  return (b & 0x80000000u) ? ~b : (b | 0x80000000u);
}
__device__ __forceinline__ float dec_f32(unsigned e) {
  unsigned b = (e & 0x80000000u) ? (e & 0x7FFFFFFFu) : ~e;
  return __uint_as_float(b);
}

__global__ void scores_kernel(const float* __restrict__ Q, const float* __restrict__ K,
                              const int* __restrict__ src, const int* __restrict__ dst,
                              float* __restrict__ scores, unsigned* __restrict__ mEnc) {
  int e = blockIdx.x * blockDim.x + threadIdx.x;
  if (e >= N_EDGES) return;
  int s = src[e], d = dst[e];
  const float4* qp = (const float4*)(Q + (size_t)d * DIM);
  const float4* kp = (const float4*)(K + (size_t)s * DIM);
  float acc = 0.f;
#pragma unroll
  for (int i = 0; i < 16; ++i) {
    float4 q = qp[i], k = kp[i];
    acc += q.x * k.x + q.y * k.y + q.z * k.z + q.w * k.w;
  }
  acc *= 0.125f;   // 1/sqrt(64)
  scores[e] = acc;
  atomicMax(&mEnc[d], enc_f32(acc));
}

__global__ void exp_kernel(float* __restrict__ scores, const unsigned* __restrict__ mEnc,
                           const int* __restrict__ dst, float* __restrict__ denom) {
  int e = blockIdx.x * blockDim.x + threadIdx.x;
  if (e >= N_EDGES) return;
  int d = dst[e];
  float m = dec_f32(mEnc[d]);   // finite: this dst has >=1 edge
  float ev = expf(scores[e] - m);
  scores[e] = ev;
  atomicAdd(&denom[d], ev);
}

// One wave per edge: lane handles feature dims {lane, lane+32}.
__global__ void agg_kernel(const float* __restrict__ escore, const float* __restrict__ denom,
                           const float* __restrict__ V,
                           const int* __restrict__ src, const int* __restrict__ dst,
                           float* __restrict__ T) {
  int wid = (blockIdx.x * blockDim.x + threadIdx.x) >> 5;
  int lane = threadIdx.x & 31;
  if (wid >= N_EDGES) return;
  int s = src[wid], d = dst[wid];
  float alpha = escore[wid] / (denom[d] + 1e-16f);
  const float* vp = V + (size_t)s * DIM;
  float* tp = T + (size_t)d * DIM;
  atomicAdd(&tp[lane],      alpha * vp[lane]);
  atomicAdd(&tp[lane + 32], alpha * vp[lane + 32]);
}

// ---------------------------------------------------------------------------
// Elementwise combines
// ---------------------------------------------------------------------------
__global__ void add_sigmoid_kernel(const float* __restrict__ A, const float* __restrict__ B,
                                   float* __restrict__ O, int n) {
  int i = blockIdx.x * blockDim.x + threadIdx.x;
  if (i < n) { float v = A[i] + B[i]; O[i] = 1.f / (1.f + expf(-v)); }
}

__global__ void mul_inplace_kernel(float* __restrict__ R, const float* __restrict__ H, int n) {
  int i = blockIdx.x * blockDim.x + threadIdx.x;
  if (i < n) R[i] = R[i] * H[i];
}

__global__ void final_kernel(const float* __restrict__ Z, const float* __restrict__ H,
                             const float* __restrict__ A, const float* __restrict__ B,
                             float* __restrict__ out, int n) {
  int i = blockIdx.x * blockDim.x + threadIdx.x;
  if (i < n) {
    float z = Z[i];
    float ht = tanhf(A[i] + B[i]);
    out[i] = z * H[i] + (1.f - z) * ht;
  }
}

// ---------------------------------------------------------------------------
// Host-side orchestration
// ---------------------------------------------------------------------------
extern "C" void kernel_launch(void* const* d_in, const int* in_sizes, int n_in,
                              void* d_out, int out_size, void* d_ws, size_t ws_size,
                              hipStream_t stream) {
  (void)in_sizes; (void)n_in; (void)out_size; (void)ws_size;
  const float* x  = (const float*)d_in[0];
  const float* h  = (const float*)d_in[1];
  const int*   ei = (const int*)d_in[2];
  const float* Wq = (const float*)d_in[3];
  const float* bq = (const float*)d_in[4];
  const float* Wk = (const float*)d_in[5];
  const float* bk = (const float*)d_in[6];
  const float* Wv = (const float*)d_in[7];
  const float* bv = (const float*)d_in[8];
  const float* Ws = (const float*)d_in[9];
  const float* bs = (const float*)d_in[10];
  const int* src = ei;
  const int* dst = ei + N_EDGES;

  float* ws = (float*)d_ws;
  float* Q      = ws + (size_t)0 * ND;
  float* K      = ws + (size_t)1 * ND;
  float* V      = ws + (size_t)2 * ND;
  float* T0     = ws + (size_t)3 * ND;
  float* T1     = ws + (size_t)4 * ND;
  float* Z      = ws + (size_t)5 * ND;
  float* R      = ws + (size_t)6 * ND;
  float* scores = ws + (size_t)7 * ND;
  unsigned* mEnc = (unsigned*)(scores + N_EDGES);
  float* denom   = (float*)(mEnc + N_NODES);

  dim3 gemm_grid((MTILES + 7) / 8, 4);
  int eb = (N_EDGES + 255) / 256;
  int nb = (N_NODES + 255) / 256;
  int ndb = (ND + 255) / 256;
  int aggb = (N_EDGES * 32 + 255) / 256;

  auto run_conv = [&](const float* inp, int c, float* T) {
    gemm_qkvs_kernel<<<gemm_grid, 256, 0, stream>>>(
        inp,
        Wq + (size_t)c * DIM * DIM, bq + (size_t)c * DIM,
        Wk + (size_t)c * DIM * DIM, bk + (size_t)c * DIM,
        Wv + (size_t)c * DIM * DIM, bv + (size_t)c * DIM,
        Ws + (size_t)c * DIM * DIM, bs + (size_t)c * DIM,
        Q, K, V, T);                                   // T := inp@Ws + bs (skip)
    init_seg_kernel<<<nb, 256, 0, stream>>>(mEnc, denom, N_NODES);
    scores_kernel<<<eb, 256, 0, stream>>>(Q, K, src, dst, scores, mEnc);
    exp_kernel<<<eb, 256, 0, stream>>>(scores, mEnc, dst, denom);
    agg_kernel<<<aggb, 256, 0, stream>>>(scores, denom, V, src, dst, T);  // T += agg
  };

  // z = sigmoid(conv0(x) + conv1(h))
  run_conv(x, 0, T0);
  run_conv(h, 1, T1);
  add_sigmoid_kernel<<<ndb, 256, 0, stream>>>(T0, T1, Z, ND);
  // r = sigmoid(conv2(x) + conv3(h))
  run_conv(x, 2, T0);
  run_conv(h, 3, T1);
  add_sigmoid_kernel<<<ndb, 256, 0, stream>>>(T0, T1, R, ND);
  // h_tilde = tanh(conv4(x) + conv5(r*h))
  run_conv(x, 4, T0);
  mul_inplace_kernel<<<ndb, 256, 0, stream>>>(R, h, ND);   // R := r*h
  run_conv(R, 5, T1);
  // out = z*h + (1-z)*h_tilde
  final_kernel<<<ndb, 256, 0, stream>>>(Z, h, T0, T1, (float*)d_out, ND);
}